// GCN_60859686584882
// MI455X (gfx1250) — compile-verified
//
#include <hip/hip_runtime.h>

typedef __attribute__((ext_vector_type(2))) float v2f;
typedef __attribute__((ext_vector_type(8))) float v8f;

#define NN 100000
#define NE 1600000
#define FIN 128
#define HD 128
#define NC 40

// ---------------- degree / normalization ----------------

__global__ void k_init_deg(float* __restrict__ deg) {
    int i = blockIdx.x * blockDim.x + threadIdx.x;
    if (i < NN) deg[i] = 1.0f;                 // self-loop
}

__global__ void k_count_deg(const int* __restrict__ ei, float* __restrict__ deg) {
    int e = blockIdx.x * blockDim.x + threadIdx.x;
    if (e < NE) atomicAdd(&deg[ei[NE + e]], 1.0f);   // dst row
}

__global__ void k_deg_to_dinv(float* __restrict__ deg) {
    int i = blockIdx.x * blockDim.x + threadIdx.x;
    if (i < NN) deg[i] = rsqrtf(deg[i]);
}

__global__ void k_zero(float* __restrict__ p, long n) {
    long i = (long)blockIdx.x * blockDim.x + threadIdx.x;
    if (i < n) p[i] = 0.0f;
}

// ---------------- GEMM1: h = x @ W1  (100000x128 @ 128x128), f32 WMMA ----------------
// block = 256 threads = 8 waves; block b handles rows [16b,16b+16); wave w -> cols [16w,16w+16)

__global__ __launch_bounds__(256) void k_gemm1(const float* __restrict__ x,
                                               const float* __restrict__ W,
                                               float* __restrict__ h) {
    const int lane   = threadIdx.x & 31;
    const int wave   = threadIdx.x >> 5;
    const int mn     = lane & 15;          // A: M index, B: N index
    const int khalf  = lane >> 4;          // selects K pair {0,1} vs {2,3}
    const int rowB   = blockIdx.x * 16;
    const int colB   = wave * 16;

    const float* arow = x + (size_t)(rowB + mn) * FIN + 2 * khalf;
    const float* bcol = W + (size_t)(2 * khalf) * HD + colB + mn;

    v8f acc = {};
#pragma unroll
    for (int k4 = 0; k4 < FIN / 4; ++k4) {
        v2f a = *(const v2f*)(arow + 4 * k4);                // K = 4k+2kh, 4k+2kh+1
        v2f b;
        b.x = bcol[(size_t)(4 * k4) * HD];
        b.y = bcol[(size_t)(4 * k4 + 1) * HD];
        acc = __builtin_amdgcn_wmma_f32_16x16x4_f32(false, a, false, b,
                                                    (short)0, acc, false, false);
    }
    // D: vgpr r -> row rowB + 8*khalf + r, col colB + mn
    float* o = h + (size_t)(rowB + 8 * khalf) * HD + colB + mn;
#pragma unroll
    for (int r = 0; r < 8; ++r) o[(size_t)r * HD] = acc[r];
}

// ---------------- edge scatter, layer 1 (width 128): wave per edge ----------------

__global__ __launch_bounds__(256) void k_scatter1(const int* __restrict__ ei,
                                                  const float* __restrict__ dinv,
                                                  const float* __restrict__ h,
                                                  float* __restrict__ agg) {
    long gid = (long)blockIdx.x * blockDim.x + threadIdx.x;
    long e   = gid >> 5;
    int lane = (int)(gid & 31);
    if (e >= NE) return;
    int src = ei[e], dst = ei[NE + e];
    float norm = dinv[src] * dinv[dst];
    const float4 v = *(const float4*)(h + (size_t)src * HD + lane * 4);
    float* o = agg + (size_t)dst * HD + lane * 4;
    atomicAdd(o + 0, norm * v.x);
    atomicAdd(o + 1, norm * v.y);
    atomicAdd(o + 2, norm * v.z);
    atomicAdd(o + 3, norm * v.w);
}

// ---------------- self-loop + bias + ReLU (in place into agg -> h1) ----------------

__global__ void k_finalize1(const float* __restrict__ h, const float* __restrict__ dinv,
                            const float* __restrict__ b1, float* __restrict__ agg) {
    long i = (long)blockIdx.x * blockDim.x + threadIdx.x;
    if (i >= (long)NN * HD) return;
    int node = (int)(i >> 7);
    int f    = (int)(i & 127);
    float d  = dinv[node];
    float v  = agg[i] + d * d * h[i] + b1[f];
    agg[i] = fmaxf(v, 0.0f);
}

// ---------------- GEMM2: z = h1 @ W2 (128 -> 40), 3 col tiles, masked ----------------

__global__ __launch_bounds__(96) void k_gemm2(const float* __restrict__ h1,
                                              const float* __restrict__ W,
                                              float* __restrict__ z) {
    const int lane  = threadIdx.x & 31;
    const int wave  = threadIdx.x >> 5;      // 0..2
    const int mn    = lane & 15;
    const int khalf = lane >> 4;
    const int rowB  = blockIdx.x * 16;
    const int colB  = wave * 16;
    const int col   = colB + mn;
    const bool cv   = (col < NC);

    const float* arow = h1 + (size_t)(rowB + mn) * HD + 2 * khalf;
    v8f acc = {};
#pragma unroll
    for (int k4 = 0; k4 < HD / 4; ++k4) {
        v2f a = *(const v2f*)(arow + 4 * k4);
        int k0 = 4 * k4 + 2 * khalf;
        v2f b;
        b.x = cv ? W[(size_t)k0 * NC + col] : 0.0f;
        b.y = cv ? W[(size_t)(k0 + 1) * NC + col] : 0.0f;
        acc = __builtin_amdgcn_wmma_f32_16x16x4_f32(false, a, false, b,
                                                    (short)0, acc, false, false);
    }
    if (cv) {
        float* o = z + (size_t)(rowB + 8 * khalf) * NC + col;
#pragma unroll
        for (int r = 0; r < 8; ++r) o[(size_t)r * NC] = acc[r];
    }
}

// ---------------- edge scatter, layer 2 (width 40) ----------------

__global__ __launch_bounds__(256) void k_scatter2(const int* __restrict__ ei,
                                                  const float* __restrict__ dinv,
                                                  const float* __restrict__ z,
                                                  float* __restrict__ out) {
    long gid = (long)blockIdx.x * blockDim.x + threadIdx.x;
    long e   = gid >> 5;
    int lane = (int)(gid & 31);
    if (e >= NE) return;
    int src = ei[e], dst = ei[NE + e];
    float norm = dinv[src] * dinv[dst];
    const float* zs = z + (size_t)src * NC;
    float* o = out + (size_t)dst * NC;
    atomicAdd(o + lane, norm * zs[lane]);
    if (lane < NC - 32) atomicAdd(o + 32 + lane, norm * zs[32 + lane]);
}

__global__ void k_finalize2(const float* __restrict__ z, const float* __restrict__ dinv,
                            const float* __restrict__ b2, float* __restrict__ out) {
    long i = (long)blockIdx.x * blockDim.x + threadIdx.x;
    if (i >= (long)NN * NC) return;
    int node = (int)(i / NC);
    int f    = (int)(i - (long)node * NC);
    float d  = dinv[node];
    out[i] += d * d * z[i] + b2[f];
}

// ---------------- launch ----------------

extern "C" void kernel_launch(void* const* d_in, const int* in_sizes, int n_in,
                              void* d_out, int out_size, void* d_ws, size_t ws_size,
                              hipStream_t stream) {
    const float* x  = (const float*)d_in[0];
    const int*   ei = (const int*)d_in[1];
    const float* W1 = (const float*)d_in[2];
    const float* b1 = (const float*)d_in[3];
    const float* W2 = (const float*)d_in[4];
    const float* b2 = (const float*)d_in[5];
    float* out = (float*)d_out;

    float* ws   = (float*)d_ws;
    float* dinv = ws;                              // N          (deg, then dinv)
    float* hbuf = ws + NN;                         // N*128      (h, later z)
    float* abuf = hbuf + (size_t)NN * HD;          // N*128      (agg, later h1)

    // degree + normalization
    k_init_deg<<<(NN + 255) / 256, 256, 0, stream>>>(dinv);
    k_count_deg<<<(NE + 255) / 256, 256, 0, stream>>>(ei, dinv);
    k_deg_to_dinv<<<(NN + 255) / 256, 256, 0, stream>>>(dinv);

    // layer 1
    k_gemm1<<<NN / 16, 256, 0, stream>>>(x, W1, hbuf);
    k_zero<<<(int)(((long)NN * HD + 255) / 256), 256, 0, stream>>>(abuf, (long)NN * HD);
    k_scatter1<<<(int)(((long)NE * 32 + 255) / 256), 256, 0, stream>>>(ei, dinv, hbuf, abuf);
    k_finalize1<<<(int)(((long)NN * HD + 255) / 256), 256, 0, stream>>>(hbuf, dinv, b1, abuf);

    // layer 2 (h1 lives in abuf, z overwrites hbuf)
    k_gemm2<<<NN / 16, 96, 0, stream>>>(abuf, W2, hbuf);
    k_zero<<<(int)(((long)NN * NC + 255) / 256), 256, 0, stream>>>(out, (long)NN * NC);
    k_scatter2<<<(int)(((long)NE * 32 + 255) / 256), 256, 0, stream>>>(ei, dinv, hbuf, out);
    k_finalize2<<<(int)(((long)NN * NC + 255) / 256), 256, 0, stream>>>(hbuf, dinv, b2, out);
}